// SelfLoopLinear_15771119911597
// MI455X (gfx1250) — compile-verified
//
#include <hip/hip_runtime.h>
#include <hip/hip_bf16.h>
#include <math.h>

// ---------------------------------------------------------------------------
// SelfLoopLinear on MI455X (gfx1250, wave32).
//
// Per step: h2 = h@W^T + b ; G = h2^T h2 ; eig(G) via parallel Jacobi ;
// sort spectrum -> rank/trunc/l1/scale ; h = h2 @ (V diag(c) V^T).
// All GEMMs run on V_WMMA_F32_16X16X4_F32 (fp32 matrix pipe — the spectrum
// is too sensitive for f16/f8). Each wave register-blocks a 32x32 output
// region (2x2 WMMA tiles): 4 WMMAs per K-step share 2 A-frags + 2 B-frags,
// doubling FLOP/byte vs one-tile-per-wave and giving 4 independent
// accumulators for matrix-pipe ILP. Eigensolve + sort use LDS.
// ---------------------------------------------------------------------------

#define DIMN 512
#define BATCHN 4096
#define STEPSN 30
#define GRANK 321
#define LAMBDA0 15.045f
#define RANKEPS 1e-5f
#define NORMF 1448.1546875f   // sqrt(4096*512) = 1024*sqrt(2)

typedef __attribute__((ext_vector_type(2))) float v2f;
typedef __attribute__((ext_vector_type(8))) float v8f;

// ---------------------------------------------------------------------------
// fp32 WMMA GEMM, 32x32 per wave: C[M,N] = op(A) @ op(B) (+ bias)
//   AT:   A(m,k) = A[k*lda + m]   (A transposed, for Gram matrix)
//   BT:   B(k,n) = B[n*ldb + k]   (B transposed, for W^T and V^T)
//   CS:   A columns scaled by cscale[k]  (for V * diag(coef))
//
// Fragment layouts per ISA 7.12.2 (wave32):
//   A 16x4 (2 VGPR): lanes 0-15 -> K={0,1}, lanes 16-31 -> K={2,3}, M=lane&15
//   B 4x16 (2 VGPR): lanes 0-15 -> rows K={0,1} at N=lane, lanes 16-31 K={2,3}
//   C/D  (8 VGPR):   vgpr r, lanes 0-15 -> (M=r, N=lane); lanes 16-31 -> M=r+8
// ---------------------------------------------------------------------------
template<bool AT, bool BT, bool BIAS, bool CS>
__device__ __forceinline__ v2f load_afrag(const float* __restrict__ A, int lda,
                                          int m0, int ka,
                                          const float* __restrict__ cscale,
                                          int r)
{
    v2f a;
    if (!AT) {
        const float* ap = A + (size_t)(m0 + r) * lda + ka;
        a.x = ap[0];
        a.y = ap[1];
    } else {
        const float* ap = A + (size_t)ka * lda + (m0 + r);
        a.x = ap[0];
        a.y = ap[lda];
    }
    if (CS) {
        a.x *= cscale[ka];
        a.y *= cscale[ka + 1];
    }
    return a;
}

template<bool BT>
__device__ __forceinline__ v2f load_bfrag(const float* __restrict__ B, int ldb,
                                          int n0, int ka, int r)
{
    v2f b;
    if (BT) {
        const float* bp = B + (size_t)(n0 + r) * ldb + ka;
        b.x = bp[0];
        b.y = bp[1];
    } else {
        const float* bp = B + (size_t)ka * ldb + (n0 + r);
        b.x = bp[0];
        b.y = bp[ldb];
    }
    return b;
}

template<bool AT, bool BT, bool BIAS, bool CS>
__global__ __launch_bounds__(256)
void wmma_gemm_f32(const float* __restrict__ A, int lda,
                   const float* __restrict__ B, int ldb,
                   const float* __restrict__ bias,
                   const float* __restrict__ cscale,
                   float* __restrict__ C, int ldc,
                   int Mdim, int Ndim, int Kdim)
{
    const int lane  = threadIdx.x & 31;
    const int wave  = threadIdx.x >> 5;
    const int tilesN = Ndim >> 5;                 // 32-wide macro tiles
    const int tiles  = (Mdim >> 5) * tilesN;
    const int tile   = blockIdx.x * (blockDim.x >> 5) + wave;
    if (tile >= tiles) return;

    const int m0   = (tile / tilesN) << 5;        // 32-row block
    const int n0   = (tile % tilesN) << 5;        // 32-col block
    const int r    = lane & 15;
    const int half = lane >> 4;                   // 0: K pair {0,1}, 1: {2,3}

    v8f acc00, acc01, acc10, acc11;
    {
        float b0 = BIAS ? bias[n0 + r] : 0.0f;
        float b1 = BIAS ? bias[n0 + 16 + r] : 0.0f;
#pragma unroll
        for (int i = 0; i < 8; ++i) {
            acc00[i] = b0; acc01[i] = b1;
            acc10[i] = b0; acc11[i] = b1;
        }
    }

#pragma unroll 4
    for (int k0 = 0; k0 < Kdim; k0 += 4) {
        const int ka = k0 + 2 * half;

        if ((k0 & 63) == 0) {
            if (!AT) __builtin_prefetch(A + (size_t)(m0 + r) * lda + ka + 64, 0, 3);
            else     __builtin_prefetch(A + (size_t)(ka + 4) * lda + (m0 + r), 0, 3);
            if (BT)  __builtin_prefetch(B + (size_t)(n0 + r) * ldb + ka + 64, 0, 3);
            else     __builtin_prefetch(B + (size_t)(ka + 4) * ldb + (n0 + r), 0, 3);
        }

        v2f a0 = load_afrag<AT, BT, BIAS, CS>(A, lda, m0,      ka, cscale, r);
        v2f a1 = load_afrag<AT, BT, BIAS, CS>(A, lda, m0 + 16, ka, cscale, r);
        v2f b0 = load_bfrag<BT>(B, ldb, n0,      ka, r);
        v2f b1 = load_bfrag<BT>(B, ldb, n0 + 16, ka, r);

        // 4 independent accumulators -> ILP on the f32 matrix pipe
        acc00 = __builtin_amdgcn_wmma_f32_16x16x4_f32(
                    false, a0, false, b0, (short)0, acc00, false, false);
        acc01 = __builtin_amdgcn_wmma_f32_16x16x4_f32(
                    false, a0, false, b1, (short)0, acc01, false, false);
        acc10 = __builtin_amdgcn_wmma_f32_16x16x4_f32(
                    false, a1, false, b0, (short)0, acc10, false, false);
        acc11 = __builtin_amdgcn_wmma_f32_16x16x4_f32(
                    false, a1, false, b1, (short)0, acc11, false, false);
    }

#pragma unroll
    for (int vr = 0; vr < 8; ++vr) {
        const int row0 = m0 + vr + 8 * half;
        C[(size_t)row0        * ldc + (n0 + r)]      = acc00[vr];
        C[(size_t)row0        * ldc + (n0 + 16 + r)] = acc01[vr];
        C[(size_t)(row0 + 16) * ldc + (n0 + r)]      = acc10[vr];
        C[(size_t)(row0 + 16) * ldc + (n0 + 16 + r)] = acc11[vr];
    }
}

// ---------------------------------------------------------------------------
__global__ void set_identity(float* __restrict__ V, int n)
{
    int i = blockIdx.x * blockDim.x + threadIdx.x;
    if (i < n * n) V[i] = ((i / n) == (i % n)) ? 1.0f : 0.0f;
}

// ---------------------------------------------------------------------------
// Parallel-ordered two-sided Jacobi eigensolver for symmetric 512x512 G.
// One 1024-thread workgroup; tournament (circle-method) pairing gives 256
// disjoint (p,q) pairs per round, 511 rounds per sweep. V accumulates
// eigenvectors (columns). Rotation params staged in LDS.
// ---------------------------------------------------------------------------
__global__ __launch_bounds__(1024)
void jacobi_eig(float* __restrict__ G, float* __restrict__ V,
                float* __restrict__ evals, int nsweeps)
{
    __shared__ float lc[256];
    __shared__ float ls[256];
    __shared__ int   lp[256];
    __shared__ int   lq[256];
    const int tid = threadIdx.x;
    const int N = DIMN, NR = DIMN - 1;   // 511 rounds

    for (int sweep = 0; sweep < nsweeps; ++sweep) {
        for (int rnd = 0; rnd < NR; ++rnd) {
            if (tid < 256) {
                int p, q;
                if (tid == 0) { p = N - 1; q = rnd % NR; }
                else {
                    p = (rnd + tid) % NR;
                    q = (rnd + NR - tid) % NR;
                }
                float app = G[(size_t)p * N + p];
                float aqq = G[(size_t)q * N + q];
                float apq = G[(size_t)p * N + q];
                float c = 1.0f, s = 0.0f;
                if (fabsf(apq) > 1e-30f) {
                    float tau = (aqq - app) / (2.0f * apq);
                    float t = (tau >= 0.0f ? 1.0f : -1.0f) /
                              (fabsf(tau) + sqrtf(1.0f + tau * tau));
                    c = 1.0f / sqrtf(1.0f + t * t);
                    s = t * c;
                }
                lc[tid] = c; ls[tid] = s; lp[tid] = p; lq[tid] = q;
            }
            __syncthreads();

            // Row phase: G <- J^T G   (rows p,q of all 256 disjoint pairs)
            for (int idx = tid; idx < 256 * N; idx += 1024) {
                int pr = idx >> 9, j = idx & (N - 1);
                int p = lp[pr], q = lq[pr];
                float c = lc[pr], s = ls[pr];
                float gp = G[(size_t)p * N + j];
                float gq = G[(size_t)q * N + j];
                G[(size_t)p * N + j] = c * gp - s * gq;
                G[(size_t)q * N + j] = s * gp + c * gq;
            }
            __syncthreads();

            // Column phase: G <- G J ; accumulate V <- V J
            for (int idx = tid; idx < 256 * N; idx += 1024) {
                int pr = idx >> 9, i = idx & (N - 1);
                int p = lp[pr], q = lq[pr];
                float c = lc[pr], s = ls[pr];
                float gp = G[(size_t)i * N + p];
                float gq = G[(size_t)i * N + q];
                G[(size_t)i * N + p] = c * gp - s * gq;
                G[(size_t)i * N + q] = s * gp + c * gq;
                float vp = V[(size_t)i * N + p];
                float vq = V[(size_t)i * N + q];
                V[(size_t)i * N + p] = c * vp - s * vq;
                V[(size_t)i * N + q] = s * vp + c * vq;
            }
            __syncthreads();
        }
    }
    if (tid < N) evals[tid] = G[(size_t)tid * N + tid];
}

// ---------------------------------------------------------------------------
// Spectrum kernel: bitonic-sort eigenvalues (descending) with index payload,
// compute rank, truncation, l1 = sum(sqrt(lambda))/NORM, rescale factor,
// and emit per-eigendirection keep coefficients c[origIdx] in {scale, 0}.
// ---------------------------------------------------------------------------
__global__ __launch_bounds__(512)
void spectrum_kernel(const float* __restrict__ evals,
                     float* __restrict__ coef,
                     float* __restrict__ out_l1)
{
    __shared__ float key[DIMN];
    __shared__ int   sidx[DIMN];
    __shared__ float red[DIMN];
    __shared__ int   srank;
    __shared__ float ssum;
    const int tid = threadIdx.x;

    float v = evals[tid];
    key[tid]  = v > 0.0f ? v : 0.0f;   // clamp numeric negatives
    sidx[tid] = tid;
    __syncthreads();

    // Bitonic sort, descending by key.
    for (int k = 2; k <= DIMN; k <<= 1) {
        for (int j = k >> 1; j > 0; j >>= 1) {
            int ixj = tid ^ j;
            if (ixj > tid) {
                bool up = ((tid & k) == 0);       // descending blocks
                float a = key[tid], b = key[ixj];
                bool doswap = up ? (a < b) : (a > b);
                if (doswap) {
                    key[tid] = b; key[ixj] = a;
                    int t = sidx[tid]; sidx[tid] = sidx[ixj]; sidx[ixj] = t;
                }
            }
            __syncthreads();
        }
    }

    // current_rank = #{ sqrt(lambda) > eps }  <=>  lambda > eps^2
    red[tid] = (key[tid] > RANKEPS * RANKEPS) ? 1.0f : 0.0f;
    __syncthreads();
    for (int off = DIMN / 2; off > 0; off >>= 1) {
        if (tid < off) red[tid] += red[tid + off];
        __syncthreads();
    }
    if (tid == 0) srank = (int)(red[0] + 0.5f);
    __syncthreads();

    const bool trunc = srank > GRANK;
    const int keepN  = trunc ? GRANK : DIMN;
    const bool kept  = tid < keepN;

    red[tid] = kept ? sqrtf(key[tid]) : 0.0f;   // sum of kept singular values
    __syncthreads();
    for (int off = DIMN / 2; off > 0; off >>= 1) {
        if (tid < off) red[tid] += red[tid + off];
        __syncthreads();
    }
    if (tid == 0) ssum = red[0];
    __syncthreads();

    const float sum = ssum;
    const float l1  = sum * (1.0f / NORMF);
    const float scale = (l1 > LAMBDA0) ? (LAMBDA0 * NORMF / sum) : 1.0f;

    coef[sidx[tid]] = kept ? scale : 0.0f;
    if (tid == 0) out_l1[0] = l1;
}

// ---------------------------------------------------------------------------
extern "C" void kernel_launch(void* const* d_in, const int* in_sizes, int n_in,
                              void* d_out, int out_size, void* d_ws, size_t ws_size,
                              hipStream_t stream)
{
    const float* x = (const float*)d_in[0];   // [4096, 512]
    const float* W = (const float*)d_in[1];   // [512, 512]
    const float* b = (const float*)d_in[2];   // [512]
    float* out = (float*)d_out;               // [30] l1 sums

    char* ws = (char*)d_ws;
    float* h     = (float*)ws;  ws += (size_t)BATCHN * DIMN * sizeof(float);
    float* h2    = (float*)ws;  ws += (size_t)BATCHN * DIMN * sizeof(float);
    float* G     = (float*)ws;  ws += (size_t)DIMN * DIMN * sizeof(float);
    float* V     = (float*)ws;  ws += (size_t)DIMN * DIMN * sizeof(float);
    float* Mm    = (float*)ws;  ws += (size_t)DIMN * DIMN * sizeof(float);
    float* evals = (float*)ws;  ws += (size_t)DIMN * sizeof(float);
    float* coef  = (float*)ws;  ws += (size_t)DIMN * sizeof(float);

    const int tilesBig   = (BATCHN / 32) * (DIMN / 32);   // 2048 macro tiles
    const int tilesSmall = (DIMN / 32) * (DIMN / 32);     // 256 macro tiles
    const int blkBig   = (tilesBig + 7) / 8;              // 8 waves/block
    const int blkSmall = (tilesSmall + 7) / 8;

    for (int step = 0; step < STEPSN; ++step) {
        const float* hin = (step == 0) ? x : h;

        // h2 = hin @ W^T + b      [4096,512] x [512,512]^T
        wmma_gemm_f32<false, true, true, false>
            <<<blkBig, 256, 0, stream>>>(hin, DIMN, W, DIMN, b, nullptr,
                                         h2, DIMN, BATCHN, DIMN, DIMN);

        // G = h2^T @ h2           [512,512], K = 4096
        wmma_gemm_f32<true, false, false, false>
            <<<blkSmall, 256, 0, stream>>>(h2, DIMN, h2, DIMN, nullptr, nullptr,
                                           G, DIMN, DIMN, DIMN, BATCHN);

        // V = I, then Jacobi eigendecomposition of G
        set_identity<<<(DIMN * DIMN + 255) / 256, 256, 0, stream>>>(V, DIMN);
        jacobi_eig<<<1, 1024, 0, stream>>>(G, V, evals, 10);

        // Sort spectrum, rank/trunc/l1/scale; keep-coefficients -> coef
        spectrum_kernel<<<1, DIMN, 0, stream>>>(evals, coef, out + step);

        // Mm = (V * diag(coef)) @ V^T
        wmma_gemm_f32<false, true, false, true>
            <<<blkSmall, 256, 0, stream>>>(V, DIMN, V, DIMN, nullptr, coef,
                                           Mm, DIMN, DIMN, DIMN, DIMN);

        // h = h2 @ Mm             (truncated + rescaled reconstruction)
        wmma_gemm_f32<false, false, false, false>
            <<<blkBig, 256, 0, stream>>>(h2, DIMN, Mm, DIMN, nullptr, nullptr,
                                         h, DIMN, BATCHN, DIMN, DIMN);
    }
}